// OuterProductNN_k4_25108378812918
// MI455X (gfx1250) — compile-verified
//
#include <hip/hip_runtime.h>
#include <cmath>

// CDNA5 wave32 ext-vector fragment types
typedef __attribute__((ext_vector_type(2))) float v2f;   // A/B frag of V_WMMA_F32_16X16X4_F32
typedef __attribute__((ext_vector_type(8))) float v8f;   // C/D frag (16x16 f32)

#define NBATCH 32768
#define ROWS_PER_WAVE 16
#define WAVES_PER_BLOCK 8
#define ROWS_PER_BLOCK (ROWS_PER_WAVE * WAVES_PER_BLOCK)   // 128

// out[b] = log( R_b^T W R_b - I_b^T W I_b ),  R/I = real/imag of z (x) conj(z), d=8.
__global__ __launch_bounds__(256) void bihom_k4_wmma_kernel(
    const float* __restrict__ zr, const float* __restrict__ zi,
    const float* __restrict__ w,  float* __restrict__ out)
{
  __shared__ float sW[64 * 64];                       // 16 KB: W[p*64+q]
  __shared__ float sT[WAVES_PER_BLOCK * 16 * 64];     // 32 KB: per-wave 16x64 T tile (reused R then I)

  const int tid  = threadIdx.x;
  const int wave = tid >> 5;
  const int lane = tid & 31;
  const int m    = lane & 15;       // matrix row within the 16-row tile
  const int half = lane >> 4;       // 0: lanes 0-15, 1: lanes 16-31

  // ---- stage W into LDS: 4096 floats, 16 per thread (4x float4) ----
  {
    const float4* wsrc = reinterpret_cast<const float4*>(w);
    float4*       wdst = reinterpret_cast<float4*>(sW);
#pragma unroll
    for (int k = 0; k < 4; ++k)
      wdst[tid * 4 + k] = wsrc[tid * 4 + k];
  }

  // ---- per-lane z row (lanes 0-15 and 16-31 duplicate the 16 rows) ----
  const int row = blockIdx.x * ROWS_PER_BLOCK + wave * ROWS_PER_WAVE + m;
  float ra[8], rb[8];
  {
    const float4* zr4 = reinterpret_cast<const float4*>(zr + row * 8);
    const float4* zi4 = reinterpret_cast<const float4*>(zi + row * 8);
    float4 r0 = zr4[0], r1 = zr4[1];
    float4 i0 = zi4[0], i1 = zi4[1];
    ra[0] = r0.x; ra[1] = r0.y; ra[2] = r0.z; ra[3] = r0.w;
    ra[4] = r1.x; ra[5] = r1.y; ra[6] = r1.z; ra[7] = r1.w;
    rb[0] = i0.x; rb[1] = i0.y; rb[2] = i0.z; rb[3] = i0.w;
    rb[4] = i1.x; rb[5] = i1.y; rb[6] = i1.z; rb[7] = i1.w;
  }

  __syncthreads();   // sW ready

  // ---- T_R = Rtile(16x64) @ W, T_I = Itile(16x64) @ W via f32 WMMA, K in 16 chunks of 4 ----
  v8f accR[4], accI[4];
#pragma unroll
  for (int n = 0; n < 4; ++n) { accR[n] = {}; accI[n] = {}; }

#pragma unroll
  for (int c = 0; c < 16; ++c) {
    // p = 4c + 2*half + v  ->  i = p>>3 = c>>1 (const), j = 4*(c&1) + 2*half + v
    const int  i  = c >> 1;
    const int  jb = (c & 1) * 4;
    v2f aR, aI;
#pragma unroll
    for (int v = 0; v < 2; ++v) {
      const float aj = half ? ra[jb + 2 + v] : ra[jb + v];
      const float bj = half ? rb[jb + 2 + v] : rb[jb + v];
      aR[v] = ra[i] * aj + rb[i] * bj;      // R[m, p]
      aI[v] = rb[i] * aj - ra[i] * bj;      // I[m, p]
    }
    const int k0 = 4 * c + 2 * half;        // B frag: W[k0+v, 16n+m]
#pragma unroll
    for (int n = 0; n < 4; ++n) {
      v2f bw;
      bw[0] = sW[(k0 + 0) * 64 + 16 * n + m];
      bw[1] = sW[(k0 + 1) * 64 + 16 * n + m];
      accR[n] = __builtin_amdgcn_wmma_f32_16x16x4_f32(
          false, aR, false, bw, (short)0, accR[n], false, false);
      accI[n] = __builtin_amdgcn_wmma_f32_16x16x4_f32(
          false, aI, false, bw, (short)0, accI[n], false, false);
    }
  }

  float* tbuf = &sT[wave * 16 * 64];

  // ---- x_R = rowwise dot(T_R, R) via LDS bounce ----
#pragma unroll
  for (int n = 0; n < 4; ++n)
#pragma unroll
    for (int v = 0; v < 8; ++v)
      tbuf[(v + 8 * half) * 64 + 16 * n + m] = accR[n][v];   // C-layout: row v+8*half, col 16n+m
  __syncthreads();

  float xR = 0.f;
#pragma unroll
  for (int q = 0; q < 64; ++q) {
    const int i = q >> 3, j = q & 7;
    xR += tbuf[m * 64 + q] * (ra[i] * ra[j] + rb[i] * rb[j]);
  }
  __syncthreads();   // all reads of tbuf done before overwrite

  // ---- x_I = rowwise dot(T_I, I) ----
#pragma unroll
  for (int n = 0; n < 4; ++n)
#pragma unroll
    for (int v = 0; v < 8; ++v)
      tbuf[(v + 8 * half) * 64 + 16 * n + m] = accI[n][v];
  __syncthreads();

  float xI = 0.f;
#pragma unroll
  for (int q = 0; q < 64; ++q) {
    const int i = q >> 3, j = q & 7;
    xI += tbuf[m * 64 + q] * (rb[i] * ra[j] - ra[i] * rb[j]);
  }

  if (half == 0)
    out[row] = logf(xR - xI);
}

extern "C" void kernel_launch(void* const* d_in, const int* in_sizes, int n_in,
                              void* d_out, int out_size, void* d_ws, size_t ws_size,
                              hipStream_t stream) {
  const float* zr = (const float*)d_in[0];   // z_real [32768, 8]
  const float* zi = (const float*)d_in[1];   // z_imag [32768, 8]
  const float* w  = (const float*)d_in[2];   // w      [4096, 1]
  float* out = (float*)d_out;                // [32768, 1]

  dim3 grid(NBATCH / ROWS_PER_BLOCK);        // 256 blocks
  dim3 block(256);                           // 8 wave32s
  hipLaunchKernelGGL(bihom_k4_wmma_kernel, grid, block, 0, stream, zr, zi, w, out);
}